// keypoint_embedding_2_21328807592435
// MI455X (gfx1250) — compile-verified
//
#include <hip/hip_runtime.h>

typedef __attribute__((ext_vector_type(2))) float v2f;
typedef __attribute__((ext_vector_type(8))) float v8f;

// ---------------------------------------------------------------------------
// Kernel 1: fused coord/pos embedding
//   emb0[m][o] = relu(coord . e1_w[o] + e1_b[o]) + relu(pos . e2_w[o] + e2_b[o])
//   m in [0,1024), o in [0,128). K is 3/13 -> plain VALU is the right tool.
// ---------------------------------------------------------------------------
__global__ void emb0_kernel(const float* __restrict__ x,
                            const float* __restrict__ e1w, const float* __restrict__ e1b,
                            const float* __restrict__ e2w, const float* __restrict__ e2b,
                            float* __restrict__ out) {
    int t = blockIdx.x * blockDim.x + threadIdx.x;     // t = m*128 + o
    if (t >= 1024 * 128) return;
    int m = t >> 7;
    int o = t & 127;
    const float* xr = x + m * 16;
    float s1 = e1b[o];
#pragma unroll
    for (int c = 0; c < 3; ++c) s1 += xr[c] * e1w[o * 3 + c];
    float s2 = e2b[o];
#pragma unroll
    for (int c = 0; c < 13; ++c) s2 += xr[3 + c] * e2w[o * 13 + c];
    out[t] = fmaxf(s1, 0.f) + fmaxf(s2, 0.f);
}

// ---------------------------------------------------------------------------
// Kernel 2: generic f32 GEMM via V_WMMA_F32_16X16X4_F32 (exact f32 math).
//   out[M x N] = opt_relu( act[M x K] * wgt[N x K]^T + bias[N] )
//
//   Register tiling: one wave computes a 16x64 strip (1 M-tile x 4 N-tiles).
//   Per K-step of 4: 1 A load + 4 B loads feed 4 WMMAs (1.25 loads/WMMA,
//   4x less A traffic than 1 tile/wave). 32 accumulator VGPRs per wave.
//   Requires M % 16 == 0 and N % 64 == 0 (true for every GEMM here).
//
//   A 16x4 lane layout (ISA 7.12.2): lanes 0-15: v0=K0, v1=K1 of row M=lane;
//   lanes 16-31: v0=K2, v1=K3 of row M=lane-16.  B 4x16 mirrored.
//   C/D 16x16: VGPR r = row (r + 8*(lane>=16)), col = lane&15.
// ---------------------------------------------------------------------------
__global__ void gemm_wmma_f32(const float* __restrict__ act,
                              const float* __restrict__ wgt,
                              const float* __restrict__ bias,
                              float* __restrict__ out,
                              int M, int N, int K, int ldw, int do_relu) {
    int wave   = (blockIdx.x * blockDim.x + threadIdx.x) >> 5;  // wave-uniform
    int nsn    = N >> 6;                     // 64-wide strips along N
    int total  = (M >> 4) * nsn;
    if (wave >= total) return;               // whole-wave exit: EXEC stays all-1s
    int mt = wave / nsn;
    int ns = wave % nsn;

    int lane = threadIdx.x & 31;
    int half = lane >> 4;                    // 0: K pair {0,1}, 1: K pair {2,3}
    int lr   = lane & 15;

    const float* arow = act + (size_t)(mt * 16 + lr) * K   + half * 2;
    const float* brow = wgt + (size_t)(ns * 64 + lr) * ldw + half * 2;
    const size_t bstep = (size_t)16 * ldw;   // next 16-col group of the strip

    v8f c0 = {}, c1 = {}, c2 = {}, c3 = {};
#pragma unroll 2
    for (int k = 0; k < K; k += 4) {
        v2f a  = *(const v2f*)(arow + k);                 // global_load_b64
        v2f w0 = *(const v2f*)(brow + k);
        v2f w1 = *(const v2f*)(brow + bstep     + k);
        v2f w2 = *(const v2f*)(brow + bstep * 2 + k);
        v2f w3 = *(const v2f*)(brow + bstep * 3 + k);
        // (neg_a, A, neg_b, B, c_mod, C, reuse_a, reuse_b)
        c0 = __builtin_amdgcn_wmma_f32_16x16x4_f32(false, a, false, w0, (short)0, c0, false, false);
        c1 = __builtin_amdgcn_wmma_f32_16x16x4_f32(false, a, false, w1, (short)0, c1, false, false);
        c2 = __builtin_amdgcn_wmma_f32_16x16x4_f32(false, a, false, w2, (short)0, c2, false, false);
        c3 = __builtin_amdgcn_wmma_f32_16x16x4_f32(false, a, false, w3, (short)0, c3, false, false);
    }

    int mbase = mt * 16 + half * 8;
    v8f acc[4] = {c0, c1, c2, c3};
#pragma unroll
    for (int t = 0; t < 4; ++t) {
        int   n  = ns * 64 + t * 16 + lr;
        float bb = bias ? bias[n] : 0.f;
#pragma unroll
        for (int r = 0; r < 8; ++r) {
            float v = acc[t][r] + bb;
            if (do_relu) v = fmaxf(v, 0.f);
            out[(size_t)(mbase + r) * N + n] = v;
        }
    }
}

// ---------------------------------------------------------------------------
// Kernel 3: fused pairwise-ReLU mean (never materializes (16,64,64,512) h1):
//   m[b][c] = (1/4096) * sum_{i,j} relu(A[b,i,c] + B[b,j,c] + b1[c])
//   One thread per (b,c); A column held in 64 registers, 64 streamed B loads.
// ---------------------------------------------------------------------------
__global__ void pairmean_kernel(const float* __restrict__ A,
                                const float* __restrict__ B,
                                const float* __restrict__ b1,
                                float* __restrict__ m_out) {
    int t = blockIdx.x * blockDim.x + threadIdx.x;     // t = b*512 + c
    if (t >= 16 * 512) return;
    int b = t >> 9;
    int c = t & 511;
    const float* Ab = A + (size_t)b * 64 * 512 + c;
    const float* Bb = B + (size_t)b * 64 * 512 + c;
    float bias = b1[c];

    float av[64];
#pragma unroll
    for (int i = 0; i < 64; ++i) av[i] = Ab[(size_t)i * 512];

    float acc = 0.f;
    for (int j = 0; j < 64; ++j) {
        float bj = Bb[(size_t)j * 512] + bias;
#pragma unroll
        for (int i = 0; i < 64; ++i) acc += fmaxf(av[i] + bj, 0.f);
    }
    m_out[t] = acc * (1.0f / 4096.0f);
}

// ---------------------------------------------------------------------------
extern "C" void kernel_launch(void* const* d_in, const int* in_sizes, int n_in,
                              void* d_out, int out_size, void* d_ws, size_t ws_size,
                              hipStream_t stream) {
    const float* x   = (const float*)d_in[0];
    const float* e1w = (const float*)d_in[1];
    const float* e1b = (const float*)d_in[2];
    const float* e2w = (const float*)d_in[3];
    const float* e2b = (const float*)d_in[4];
    const float* r1w = (const float*)d_in[5];
    const float* r1b = (const float*)d_in[6];
    const float* r2w = (const float*)d_in[7];
    const float* r2b = (const float*)d_in[8];
    const float* r3w = (const float*)d_in[9];
    const float* r3b = (const float*)d_in[10];
    const float* w1  = (const float*)d_in[11];
    const float* b1  = (const float*)d_in[12];
    const float* w2  = (const float*)d_in[13];
    const float* b2  = (const float*)d_in[14];
    float* out = (float*)d_out;

    // Workspace layout (floats), ~12.1 MB total
    float* ws   = (float*)d_ws;
    float* emb0 = ws;                    // 1024 x 128
    float* emb1 = emb0 + 1024 * 128;     // 1024 x 256
    float* emb2 = emb1 + 1024 * 256;     // 1024 x 512
    float* emb3 = emb2 + 1024 * 512;     // 1024 x 1024
    float* Abuf = emb3 + 1024 * 1024;    // 1024 x 512
    float* Bbuf = Abuf + 1024 * 512;     // 1024 x 512
    float* mbuf = Bbuf + 1024 * 512;     // 16 x 512

    emb0_kernel<<<512, 256, 0, stream>>>(x, e1w, e1b, e2w, e2b, emb0);

    auto gemm = [&](const float* act, const float* wgt, const float* bias, float* o,
                    int M, int N, int K, int ldw, int relu) {
        int strips = (M / 16) * (N / 64);      // one 16x64 strip per wave
        int blocks = (strips + 7) / 8;         // 8 waves per 256-thread block
        gemm_wmma_f32<<<blocks, 256, 0, stream>>>(act, wgt, bias, o, M, N, K, ldw, relu);
    };

    gemm(emb0, r1w, r1b, emb1, 1024,  256,  128,  128, 1);   // -> relu, (1024,256)
    gemm(emb1, r2w, r2b, emb2, 1024,  512,  256,  256, 1);   // -> relu, (1024,512)
    gemm(emb2, r3w, r3b, emb3, 1024, 1024,  512,  512, 1);   // -> relu, (1024,1024)
    gemm(emb3, w1,        nullptr, Abuf, 1024, 512, 1024, 2048, 0); // A = emb @ w1[:, :1024]^T
    gemm(emb3, w1 + 1024, nullptr, Bbuf, 1024, 512, 1024, 2048, 0); // B = emb @ w1[:,1024:]^T

    pairmean_kernel<<<32, 256, 0, stream>>>(Abuf, Bbuf, b1, mbuf);  // (16,512)

    gemm(mbuf, w2, b2, out, 16, 2048, 512, 512, 0);          // out = m @ w2^T + b2
}